// DGCNN_71141838291557
// MI455X (gfx1250) — compile-verified
//
#include <hip/hip_runtime.h>
#include <hip/hip_bf16.h>
#include <math.h>
#include <stdint.h>

typedef float v2f __attribute__((ext_vector_type(2)));
typedef float v8f __attribute__((ext_vector_type(8)));

// ---------------------------------------------------------------------------
// Async global->LDS helpers (CDNA5: GLOBAL_LOAD_ASYNC_TO_LDS_*, ASYNCcnt)
// ---------------------------------------------------------------------------
__device__ __forceinline__ void async_copy_b128(uint32_t lds_off, const float* g) {
    asm volatile("global_load_async_to_lds_b128 %0, %1, off"
                 :: "v"(lds_off), "v"(g) : "memory");
}

template <int NREM>
__device__ __forceinline__ void wait_async() {
#if __has_builtin(__builtin_amdgcn_s_wait_asynccnt)
    __builtin_amdgcn_s_wait_asynccnt((short)NREM);
#else
    if constexpr (NREM == 16)     asm volatile("s_wait_asynccnt 0x10" ::: "memory");
    else if constexpr (NREM == 1) asm volatile("s_wait_asynccnt 0x1"  ::: "memory");
    else                          asm volatile("s_wait_asynccnt 0x0"  ::: "memory");
#endif
}

// Stage one 128-candidate chunk (rows of D floats) into LDS: thread t copies
// candidate row t as D/4 x b128 async transfers. Per wave: D/4 instructions.
template <int D>
__device__ __forceinline__ void stage_async(const float* __restrict__ X, int chunk,
                                            float* xbuf, int tid) {
    const float* g = X + (size_t)(chunk + tid) * D;
    uint32_t l = (uint32_t)(uintptr_t)(xbuf + (size_t)tid * D);
    #pragma unroll
    for (int r = 0; r < D / 4; ++r)
        async_copy_b128(l + r * 16, g + r * 4);
}

// ---------------------------------------------------------------------------
// Row squared-norm: sq[i] = sum_d X[i*ld+d]^2
// ---------------------------------------------------------------------------
__global__ __launch_bounds__(256, 1)
void rownorm_kernel(const float* __restrict__ X, int ld, int d,
                    float* __restrict__ sq, int N) {
    int i = blockIdx.x * blockDim.x + threadIdx.x;
    if (i >= N) return;
    float s = 0.f;
    for (int k = 0; k < d; ++k) {
        float v = X[(size_t)i * ld + k];
        s += v * v;
    }
    sq[i] = s;
}

// ---------------------------------------------------------------------------
// kNN top-20. Block = 128 threads (4 waves) = 64 query rows. Candidate chunks
// of 128 rows are double-buffered in LDS via async global->LDS copies while
// WMMA distance tiles + top-k scans run on the previous chunk.
// X must be row-padded to D (multiple of 4).
// Distances d' = sq_j - 2*dot (per-row constant sq_i doesn't change top-k).
// __launch_bounds__(128,1): trade occupancy for registers so the A-strip,
// top-k lists and WMMA accumulators stay resident (no scratch spills).
// ---------------------------------------------------------------------------
template <int D>
__global__ __launch_bounds__(128, 1)
void knn_kernel(const float* __restrict__ X,
                const float* __restrict__ sq,
                int* __restrict__ out, int N) {
    __shared__ float xb[2][128 * D];  // staged candidate chunks
    __shared__ float dist[64][129];   // stride 129 -> conflict-free col scans
    __shared__ float sqj_sh[128];

    const int tid  = threadIdx.x;
    const int wave = tid >> 5;
    const int lane = tid & 31;
    const int half = lane >> 4;
    const int m    = lane & 15;
    const int rowBase = blockIdx.x * 64;
    constexpr int KS = D / 4;

    // Cache this wave's A strip (16 query rows x D) in registers.
    v2f areg[KS];
    {
        int row = rowBase + wave * 16 + m;
        #pragma unroll
        for (int ks = 0; ks < KS; ++ks) {
            int k0 = ks * 4 + 2 * half;
            areg[ks].x = X[(size_t)row * D + k0];
            areg[ks].y = X[(size_t)row * D + k0 + 1];
        }
    }

    float bd[20]; int bi[20];
    #pragma unroll
    for (int s = 0; s < 20; ++s) { bd[s] = __builtin_inff(); bi[s] = 0; }
    float worst = __builtin_inff();

    stage_async<D>(X, 0, xb[0], tid);           // prime the pipeline

    for (int chunk = 0; chunk < N; chunk += 128) {
        const int  buf  = (chunk >> 7) & 1;
        const bool more = (chunk + 128) < N;
        if (more) stage_async<D>(X, chunk + 128, xb[buf ^ 1], tid);
        float myq = sq[chunk + tid];            // blockDim.x == 128
        if (more) wait_async<KS>();             // oldest KS (current buf) done
        else      wait_async<0>();
        sqj_sh[tid] = myq;
        __syncthreads();

        const float* xrow = xb[buf];
        #pragma unroll
        for (int tile = 0; tile < 8; ++tile) {
            const int tb = tile * 16;
            v8f acc = {0.f, 0.f, 0.f, 0.f, 0.f, 0.f, 0.f, 0.f};
            #pragma unroll
            for (int ks = 0; ks < KS; ++ks) {
                int k0 = ks * 4 + 2 * half;
                v2f b;                                   // ds_load_b64 from LDS
                b.x = xrow[(tb + m) * D + k0];
                b.y = xrow[(tb + m) * D + k0 + 1];
                acc = __builtin_amdgcn_wmma_f32_16x16x4_f32(
                        false, areg[ks], false, b, (short)0, acc, false, false);
            }
            #pragma unroll
            for (int r = 0; r < 8; ++r) {
                int rl = wave * 16 + r + 8 * half;        // C/D row layout
                dist[rl][tb + m] = sqj_sh[tb + m] - 2.0f * acc[r];
            }
        }
        __syncthreads();

        // Owner threads: sorted-insert top-20 (registers, fully unrolled).
        if (tid < 64) {
            for (int c = 0; c < 128; ++c) {
                float d = dist[tid][c];
                if (d < worst) {                   // strict '<': jax tie-break
                    bd[19] = d; bi[19] = chunk + c;
                    #pragma unroll
                    for (int s = 19; s > 0; --s) {
                        if (bd[s] < bd[s - 1]) {
                            float td = bd[s]; bd[s] = bd[s-1]; bd[s-1] = td;
                            int   ti = bi[s]; bi[s] = bi[s-1]; bi[s-1] = ti;
                        }
                    }
                    worst = bd[19];
                }
            }
        }
        __syncthreads();
    }

    if (tid < 64) {
        int q = rowBase + tid;
        #pragma unroll
        for (int s = 0; s < 20; ++s) out[q * 20 + s] = bi[s];
    }
}

// ---------------------------------------------------------------------------
// Generic f32 WMMA GEMM: C = act(A[M x Kpad] * B[Kpad x Npad] + bias),
// Kpad = 4*KSTEPS, A row-padded (lda >= Kpad). One 16x16 tile per wave.
// ---------------------------------------------------------------------------
template <int KSTEPS>
__global__ __launch_bounds__(256, 1)
void gemm16_kernel(const float* __restrict__ A, int lda,
                   const float* __restrict__ B, int ldb,
                   const float* __restrict__ bias,
                   float* __restrict__ C, int ldc,
                   int M, int Npad, int nstore, int relu) {
    const int tid  = threadIdx.x;
    const int lane = tid & 31;
    const int half = lane >> 4;
    const int m    = lane & 15;

    const int wg      = blockIdx.x * (blockDim.x >> 5) + (tid >> 5);
    const int tilesN  = Npad >> 4;
    const int rowTile = wg / tilesN;
    const int colTile = wg - rowTile * tilesN;
    if (rowTile * 16 >= M) return;        // wave-uniform exit (EXEC stays full)

    v2f areg[KSTEPS];
    {
        int row = rowTile * 16 + m;
        #pragma unroll
        for (int ks = 0; ks < KSTEPS; ++ks) {
            int k0 = ks * 4 + 2 * half;
            areg[ks].x = A[(size_t)row * lda + k0];
            areg[ks].y = A[(size_t)row * lda + k0 + 1];
        }
    }

    const int col = colTile * 16 + m;
    v8f acc = {0.f, 0.f, 0.f, 0.f, 0.f, 0.f, 0.f, 0.f};
    #pragma unroll
    for (int ks = 0; ks < KSTEPS; ++ks) {
        int k0 = ks * 4 + 2 * half;
        v2f b;
        b.x = B[(size_t)(k0    ) * ldb + col];
        b.y = B[(size_t)(k0 + 1) * ldb + col];
        acc = __builtin_amdgcn_wmma_f32_16x16x4_f32(
                false, areg[ks], false, b, (short)0, acc, false, false);
    }

    #pragma unroll
    for (int r = 0; r < 8; ++r) {
        int row = rowTile * 16 + r + 8 * half;
        if (col < nstore) {
            float v = acc[r];
            if (bias) v += bias[col];
            if (relu) v = fmaxf(v, 0.f);
            C[(size_t)row * ldc + col] = v;
        }
    }
}

// ---------------------------------------------------------------------------
// Edge-conv combine: out[i][c] = relu(P[i][c] + max_j Q[nbr(i,j)][c] + bias[c])
// PQ row layout: [P (C cols) | Q (C cols)], stride = 2C.
// ---------------------------------------------------------------------------
__global__ __launch_bounds__(256, 1)
void combine_max_kernel(const float* __restrict__ PQ, int stride2c,
                        const int* __restrict__ nbr,
                        const float* __restrict__ bias,
                        float* __restrict__ out, int N, int C) {
    int gid = blockIdx.x * blockDim.x + threadIdx.x;
    if (gid >= N * C) return;
    int i = gid / C;
    int c = gid - i * C;
    float p = PQ[(size_t)i * stride2c + c] + bias[c];
    float mx = -__builtin_inff();
    #pragma unroll
    for (int j = 0; j < 20; ++j) {
        int jj = nbr[i * 20 + j];
        mx = fmaxf(mx, PQ[(size_t)jj * stride2c + C + c]);
    }
    out[(size_t)i * C + c] = fmaxf(p + mx, 0.f);   // max_j relu == relu(max_j)
}

// ---------------------------------------------------------------------------
// Prep kernels
// ---------------------------------------------------------------------------
__global__ __launch_bounds__(256, 1)
void prep_xpad_kernel(const float* __restrict__ x, float* __restrict__ xp, int N) {
    int t = blockIdx.x * blockDim.x + threadIdx.x;   // xp: N x 4 (col 3 = 0)
    if (t >= N * 4) return;
    int i = t >> 2, c = t & 3;
    xp[t] = (c < 3) ? x[i * 3 + c] : 0.f;
}

__global__ __launch_bounds__(256, 1)
void prep_wc1_kernel(const float* __restrict__ W1, float* __restrict__ wc) {
    int t = blockIdx.x * blockDim.x + threadIdx.x;   // wc: 4 x 128
    if (t >= 4 * 128) return;
    int k = t >> 7, c = t & 127;
    float v = 0.f;
    if (k < 3)
        v = (c < 64) ? (W1[k * 64 + c] - W1[(k + 3) * 64 + c])
                     :  W1[(k + 3) * 64 + (c - 64)];
    wc[t] = v;
}

__global__ __launch_bounds__(256, 1)
void prep_wc2_kernel(const float* __restrict__ W2, float* __restrict__ wc) {
    int t = blockIdx.x * blockDim.x + threadIdx.x;   // wc: 64 x 256
    if (t >= 64 * 256) return;
    int k = t >> 8, c = t & 255;
    wc[t] = (c < 128) ? (W2[k * 128 + c] - W2[(k + 64) * 128 + c])
                      :  W2[(k + 64) * 128 + (c - 128)];
}

__global__ __launch_bounds__(256, 1)
void prep_pad_kernel(const float* __restrict__ W, float* __restrict__ wp,
                     int ncols, int npad, int K) {
    int t = blockIdx.x * blockDim.x + threadIdx.x;   // wp: K x npad
    if (t >= K * npad) return;
    int k = t / npad, c = t - k * npad;
    wp[t] = (c < ncols) ? W[k * ncols + c] : 0.f;
}

// ---------------------------------------------------------------------------
// Host launch
// ---------------------------------------------------------------------------
extern "C" void kernel_launch(void* const* d_in, const int* in_sizes, int n_in,
                              void* d_out, int out_size, void* d_ws, size_t ws_size,
                              hipStream_t stream) {
    (void)n_in; (void)out_size; (void)ws_size;
    const float* x    = (const float*)d_in[0];
    const float* W1   = (const float*)d_in[1];
    const float* b1   = (const float*)d_in[2];
    const float* W2   = (const float*)d_in[3];
    const float* b2   = (const float*)d_in[4];
    const float* fc1w = (const float*)d_in[5];
    const float* fc1b = (const float*)d_in[6];
    const float* fc2w = (const float*)d_in[7];
    const float* fc2b = (const float*)d_in[8];

    const int N      = in_sizes[0] / 3;           // 16384
    const int nclass = in_sizes[8];               // 40
    const int npad2  = ((nclass + 15) / 16) * 16; // 48

    char* ws = (char*)d_ws;
    const size_t MB = 1u << 20;
    int*   idx1 = (int*)  (ws + 0);                       // N*20*4 (1.31 MB)
    int*   idx2 = (int*)  (ws + (size_t)(1.5 * MB));      // N*20*4
    float* sq   = (float*)(ws + 3 * MB);                  // N*4
    float* wc   = (float*)(ws + 3 * MB + (1u << 16));     // up to 64 KB
    float* xp   = (float*)(ws + 3 * MB + (2u << 16));     // N*4*4 (256 KB)
    float* PQ   = (float*)(ws + 4 * MB);                  // up to 16 MB
    float* t    = PQ;                                     // reuse after PQ dead
    float* h    = (float*)(ws + 20 * MB);                 // N*64*4 (4 MB)
    float* g    = (float*)(ws + 24 * MB);                 // N*128*4 (8 MB)

    dim3 blk256(256), blk128(128);

    // Stage 0: pad x to N x 4 (unconditional vector loads everywhere after)
    prep_xpad_kernel<<<(N * 4 + 255) / 256, blk256, 0, stream>>>(x, xp, N);

    // Stage 1: kNN on x (K padded to 4)
    rownorm_kernel<<<(N + 255) / 256, blk256, 0, stream>>>(xp, 4, 4, sq, N);
    knn_kernel<4><<<N / 64, blk128, 0, stream>>>(xp, sq, idx1, N);

    // Stage 2: edge-conv1 as GEMM + gather-max
    prep_wc1_kernel<<<2, blk256, 0, stream>>>(W1, wc);
    gemm16_kernel<1><<<(N / 16) * (128 / 16) / 8, blk256, 0, stream>>>(
        xp, 4, wc, 128, nullptr, PQ, 128, N, 128, 128, 0);
    combine_max_kernel<<<(N * 64 + 255) / 256, blk256, 0, stream>>>(
        PQ, 128, idx1, b1, h, N, 64);

    // Stage 3: kNN on h (64-dim)
    rownorm_kernel<<<(N + 255) / 256, blk256, 0, stream>>>(h, 64, 64, sq, N);
    knn_kernel<64><<<N / 64, blk128, 0, stream>>>(h, sq, idx2, N);

    // Stage 4: edge-conv2 as GEMM + gather-max
    prep_wc2_kernel<<<(64 * 256) / 256, blk256, 0, stream>>>(W2, wc);
    gemm16_kernel<16><<<(N / 16) * (256 / 16) / 8, blk256, 0, stream>>>(
        h, 64, wc, 256, nullptr, PQ, 256, N, 256, 256, 0);
    combine_max_kernel<<<(N * 128 + 255) / 256, blk256, 0, stream>>>(
        PQ, 256, idx2, b2, g, N, 128);

    // Stage 5: fc1 (relu) -> t (reuses PQ region, PQ now dead)
    gemm16_kernel<32><<<(N / 16) * (128 / 16) / 8, blk256, 0, stream>>>(
        g, 128, fc1w, 128, fc1b, t, 128, N, 128, 128, 1);

    // Stage 6: fc2 -> d_out (N pad 48, store 40)
    prep_pad_kernel<<<(128 * npad2 + 255) / 256, blk256, 0, stream>>>(
        fc2w, wc, nclass, npad2, 128);
    gemm16_kernel<32><<<((N / 16) * (npad2 / 16) + 7) / 8, blk256, 0, stream>>>(
        t, 128, wc, npad2, fc2b, (float*)d_out, nclass, N, npad2, nclass, 0);
}